// TP_65670049956023
// MI455X (gfx1250) — compile-verified
//
#include <hip/hip_runtime.h>
#include <math.h>

#define D_DIM 2048
#define F_DIM 256
#define ALPHA_F 0.005f
#define EPS_F 1e-16f
#define STOP_THR_F 0.005f
#define MAX_ITER 500
#define INV_D (1.0f / 2048.0f)

typedef float v2f __attribute__((ext_vector_type(2)));
typedef float v8f __attribute__((ext_vector_type(8)));

// ---------------- wave helpers (wave32) ----------------
__device__ __forceinline__ float wave_sum(float s) {
  #pragma unroll
  for (int off = 16; off > 0; off >>= 1) s += __shfl_xor(s, off);
  return s;
}

// ---------------- init ----------------
__global__ void ot_init_kernel(float* errp, unsigned* cnt) {
  *errp = 1.0f;
  *cnt = 0u;
}

// ---------------- row stats: sq = sum(theta^2), nrm = ||emb|| ----------------
__global__ __launch_bounds__(256) void ot_rowstats_kernel(
    const float* __restrict__ theta, const float* __restrict__ emb,
    float* __restrict__ sq, float* __restrict__ nrm) {
  int t = threadIdx.x, lane = t & 31, w = t >> 5;
  int r = blockIdx.x * 8 + w;
  const float* tr = theta + (size_t)r * F_DIM;
  const float* er = emb + (size_t)r * F_DIM;
  float st = 0.f, se = 0.f;
  #pragma unroll
  for (int j = lane * 4; j < F_DIM; j += 128) {
    float4 a = *(const float4*)&tr[j];
    st += a.x * a.x + a.y * a.y + a.z * a.z + a.w * a.w;
    float4 b = *(const float4*)&er[j];
    se += b.x * b.x + b.y * b.y + b.z * b.z + b.w * b.w;
  }
  st = wave_sum(st);
  se = wave_sum(se);
  if (lane == 0) { sq[r] = st; nrm[r] = sqrtf(se); }
}

// ---------------- fused WMMA Gram -> K = exp(-alpha*C), Pnn = cos-sim ----------------
// 64x64 output tile per workgroup (8 waves), BK=32, f32 WMMA 16x16x4.
#define LDT 36  // padded LDS row stride (floats)
__global__ __launch_bounds__(256) void ot_gram_kernel(
    const float* __restrict__ theta, const float* __restrict__ emb,
    const float* __restrict__ sq, const float* __restrict__ nrm,
    float* __restrict__ Kmat, float* __restrict__ Pnn) {
  __shared__ float At[64 * LDT], Bt[64 * LDT], Ae[64 * LDT], Be[64 * LDT];
  const int nbx = D_DIM / 64;
  int bx = blockIdx.x % nbx;  // col block
  int by = blockIdx.x / nbx;  // row block
  int rowbase = by * 64, colbase = bx * 64;
  int t = threadIdx.x;
  int lane = t & 31, w = t >> 5;
  int rt = w & 3;              // row sub-tile 0..3
  int ct0 = 2 * (w >> 2);      // col sub-tiles ct0, ct0+1
  int ct1 = ct0 + 1;
  int ln = lane & 15, hi = lane >> 4;

  v8f accT0 = {}, accT1 = {}, accE0 = {}, accE1 = {};

  int lr = t >> 3;          // 0..31
  int lc = (t & 7) * 4;     // 0..28

  for (int kb = 0; kb < F_DIM; kb += 32) {
    __syncthreads();
    #pragma unroll
    for (int rr = lr; rr < 64; rr += 32) {
      *(float4*)&At[rr * LDT + lc] = *(const float4*)&theta[(size_t)(rowbase + rr) * F_DIM + kb + lc];
      *(float4*)&Bt[rr * LDT + lc] = *(const float4*)&theta[(size_t)(colbase + rr) * F_DIM + kb + lc];
      *(float4*)&Ae[rr * LDT + lc] = *(const float4*)&emb[(size_t)(rowbase + rr) * F_DIM + kb + lc];
      *(float4*)&Be[rr * LDT + lc] = *(const float4*)&emb[(size_t)(colbase + rr) * F_DIM + kb + lc];
    }
    __syncthreads();
    #pragma unroll
    for (int kk = 0; kk < 32; kk += 4) {
      int k2 = kk + 2 * hi;
      v2f aT = *(const v2f*)&At[(rt * 16 + ln) * LDT + k2];
      v2f b0 = *(const v2f*)&Bt[(ct0 * 16 + ln) * LDT + k2];
      v2f b1 = *(const v2f*)&Bt[(ct1 * 16 + ln) * LDT + k2];
      accT0 = __builtin_amdgcn_wmma_f32_16x16x4_f32(false, aT, false, b0, (short)0, accT0, false, false);
      accT1 = __builtin_amdgcn_wmma_f32_16x16x4_f32(false, aT, false, b1, (short)0, accT1, false, false);
      v2f aE = *(const v2f*)&Ae[(rt * 16 + ln) * LDT + k2];
      v2f e0 = *(const v2f*)&Be[(ct0 * 16 + ln) * LDT + k2];
      v2f e1 = *(const v2f*)&Be[(ct1 * 16 + ln) * LDT + k2];
      accE0 = __builtin_amdgcn_wmma_f32_16x16x4_f32(false, aE, false, e0, (short)0, accE0, false, false);
      accE1 = __builtin_amdgcn_wmma_f32_16x16x4_f32(false, aE, false, e1, (short)0, accE1, false, false);
    }
  }

  // Epilogue: C/D layout -> element p of acc on lane = (M = p + 8*hi, N = ln)
  #pragma unroll
  for (int p = 0; p < 8; ++p) {
    int i = rowbase + rt * 16 + p + 8 * hi;
    int j0 = colbase + ct0 * 16 + ln;
    int j1 = colbase + ct1 * 16 + ln;
    float si = sq[i], ni = nrm[i];
    float c0 = fmaxf(si + sq[j0] - 2.0f * accT0[p], 0.0f);
    float c1 = fmaxf(si + sq[j1] - 2.0f * accT1[p], 0.0f);
    Kmat[(size_t)i * D_DIM + j0] = expf(-ALPHA_F * c0);
    Kmat[(size_t)i * D_DIM + j1] = expf(-ALPHA_F * c1);
    Pnn[(size_t)i * D_DIM + j0] = accE0[p] / (ni * nrm[j0] + EPS_F);
    Pnn[(size_t)i * D_DIM + j1] = accE1[p] / (ni * nrm[j1] + EPS_F);
  }
}

// ---------------- row sums of Pnn ----------------
__global__ __launch_bounds__(256) void ot_rowsum_kernel(const float* __restrict__ Pnn,
                                                        float* __restrict__ rs) {
  int t = threadIdx.x, lane = t & 31, w = t >> 5;
  int r = blockIdx.x * 8 + w;
  const float* row = Pnn + (size_t)r * D_DIM;
  float s = 0.f;
  for (int j = lane * 4; j < D_DIM; j += 128) {
    float4 a = *(const float4*)&row[j];
    s += a.x + a.y + a.z + a.w;
  }
  s = wave_sum(s);
  if (lane == 0) rs[r] = s;
}

// ---------------- persistent Sinkhorn ----------------
__device__ __forceinline__ void grid_barrier(unsigned* cnt, int& phase, unsigned nwg) {
  phase++;
  __threadfence();       // release our writes to device scope
  __syncthreads();
  if (threadIdx.x == 0) {
    __hip_atomic_fetch_add(cnt, 1u, __ATOMIC_ACQ_REL, __HIP_MEMORY_SCOPE_AGENT);
    unsigned target = (unsigned)phase * nwg;
    while (__hip_atomic_load(cnt, __ATOMIC_ACQUIRE, __HIP_MEMORY_SCOPE_AGENT) < target)
      __builtin_amdgcn_s_sleep(8);
  }
  __syncthreads();
  __threadfence();       // acquire: invalidate stale cache before reading others' data
}

__device__ __forceinline__ void ot_loadvec(float* xs, const float* g, int t) {
  #pragma unroll
  for (int idx = t * 4; idx < D_DIM; idx += 1024)
    *(float4*)&xs[idx] = *(const float4*)&g[idx];
}

__device__ __forceinline__ float ot_wave_dot(const float* __restrict__ Krow,
                                             const float* __restrict__ xs, int lane) {
  float s = 0.f;
  #pragma unroll 4
  for (int j = lane * 4; j < D_DIM; j += 128) {
    float4 k4 = *(const float4*)&Krow[j];
    float4 x4 = *(const float4*)&xs[j];
    s += k4.x * x4.x + k4.y * x4.y + k4.z * x4.z + k4.w * x4.w;
  }
  return wave_sum(s);  // all lanes hold result
}

__global__ __launch_bounds__(256) void ot_sinkhorn_kernel(
    const float* __restrict__ Kmat, float* __restrict__ u, float* __restrict__ v,
    float* __restrict__ bbp, float* __restrict__ errp, unsigned* cnt) {
  __shared__ float xs[D_DIM];   // 8 KB staging for the dense vector
  __shared__ float red[8];
  int t = threadIdx.x, lane = t & 31, w = t >> 5;
  int r = blockIdx.x * 8 + w;            // wave-per-row, 8 rows per WG
  const unsigned nwg = gridDim.x;
  const float* Krow = Kmat + (size_t)r * D_DIM;
  int phase = 0;

  if (lane == 0) u[r] = 1.0f;
  grid_barrier(cnt, phase, nwg);

  int cpt = 0;
  float err = 1.0f;
  float vr = 1.0f;

  while (err > STOP_THR_F && cpt < MAX_ITER) {
    // v = b / (K u + eps)     (K symmetric -> K^T u == K u)
    ot_loadvec(xs, u, t);
    __syncthreads();
    float s1 = ot_wave_dot(Krow, xs, lane);
    vr = INV_D / (s1 + EPS_F);
    if (lane == 0) v[r] = vr;
    grid_barrier(cnt, phase, nwg);

    // u = a / (K v + eps)
    ot_loadvec(xs, v, t);
    __syncthreads();
    float s2 = ot_wave_dot(Krow, xs, lane);
    if (lane == 0) u[r] = INV_D / (s2 + EPS_F);
    grid_barrier(cnt, phase, nwg);

    cpt++;
    if (cpt % 50 == 1) {
      // bb = v * (K u_new);  err = sum |bb - b|
      ot_loadvec(xs, u, t);
      __syncthreads();
      float s3 = ot_wave_dot(Krow, xs, lane);
      if (lane == 0) bbp[r] = fabsf(vr * s3 - INV_D);
      grid_barrier(cnt, phase, nwg);
      if (blockIdx.x == 0) {
        float ss = 0.f;
        for (int idx = t; idx < D_DIM; idx += 256) ss += bbp[idx];
        ss = wave_sum(ss);
        if (lane == 0) red[w] = ss;
        __syncthreads();
        if (t == 0) {
          float tot = 0.f;
          #pragma unroll
          for (int q = 0; q < 8; ++q) tot += red[q];
          errp[0] = tot;
        }
        __syncthreads();
      }
      grid_barrier(cnt, phase, nwg);
      err = __hip_atomic_load(errp, __ATOMIC_RELAXED, __HIP_MEMORY_SCOPE_AGENT);
    }
  }
}

// ---------------- KL partials: per-row reductions ----------------
__global__ __launch_bounds__(256) void ot_kl_partial_kernel(
    const float* __restrict__ Kmat, const float* __restrict__ Pnn,
    const float* __restrict__ rs, const float* __restrict__ u,
    const float* __restrict__ v, float* __restrict__ partials) {
  __shared__ float redQ[8], redP[8], redK[8];
  int i = blockIdx.x;
  int t = threadIdx.x, lane = t & 31, w = t >> 5;
  const float* Krow = Kmat + (size_t)i * D_DIM;
  const float* Prow = Pnn + (size_t)i * D_DIM;
  float rsi = 1.0f / rs[i];
  float ui = u[i];
  float sQ = 0.f, sP = 0.f, sKL = 0.f;
  for (int j = t; j < D_DIM; j += 256) {
    float pij = 0.5f * (Prow[j] * rsi + Pnn[(size_t)j * D_DIM + i] / rs[j]);
    float qc = fmaxf(ui * Krow[j] * v[j], 1e-6f);
    sQ += qc;
    sP += pij;
    sKL += pij * (logf(pij) - logf(qc));
  }
  sQ = wave_sum(sQ); sP = wave_sum(sP); sKL = wave_sum(sKL);
  if (lane == 0) { redQ[w] = sQ; redP[w] = sP; redK[w] = sKL; }
  __syncthreads();
  if (t == 0) {
    float tQ = 0.f, tP = 0.f, tK = 0.f;
    #pragma unroll
    for (int q = 0; q < 8; ++q) { tQ += redQ[q]; tP += redP[q]; tK += redK[q]; }
    partials[(size_t)i * 3 + 0] = tQ;
    partials[(size_t)i * 3 + 1] = tP;
    partials[(size_t)i * 3 + 2] = tK;
  }
}

// ---------------- final reduce + combine ----------------
__global__ __launch_bounds__(256) void ot_kl_final_kernel(const float* __restrict__ partials,
                                                          float* __restrict__ out) {
  __shared__ float redQ[8], redP[8], redK[8];
  int t = threadIdx.x, lane = t & 31, w = t >> 5;
  float sQ = 0.f, sP = 0.f, sKL = 0.f;
  for (int i = t; i < D_DIM; i += 256) {
    sQ += partials[(size_t)i * 3 + 0];
    sP += partials[(size_t)i * 3 + 1];
    sKL += partials[(size_t)i * 3 + 2];
  }
  sQ = wave_sum(sQ); sP = wave_sum(sP); sKL = wave_sum(sKL);
  if (lane == 0) { redQ[w] = sQ; redP[w] = sP; redK[w] = sKL; }
  __syncthreads();
  if (t == 0) {
    float tQ = 0.f, tP = 0.f, tK = 0.f;
    #pragma unroll
    for (int q = 0; q < 8; ++q) { tQ += redQ[q]; tP += redP[q]; tK += redK[q]; }
    // kl = sum P (log P - log Qc) + log(sumQ) * sum P
    out[0] = tK + logf(tQ) * tP;   // WEIGHT_LOSS_TP == 1.0
  }
}

// ---------------- launch ----------------
extern "C" void kernel_launch(void* const* d_in, const int* in_sizes, int n_in,
                              void* d_out, int out_size, void* d_ws, size_t ws_size,
                              hipStream_t stream) {
  const float* theta = (const float*)d_in[0];
  const float* emb = (const float*)d_in[1];
  float* out = (float*)d_out;

  float* W = (float*)d_ws;
  const size_t DD = (size_t)D_DIM * D_DIM;
  float* Kmat = W;               // 16 MB
  float* Pnn = W + DD;           // 16 MB
  float* AUX = W + 2 * DD;
  float* sq = AUX;
  float* nrm = AUX + D_DIM;
  float* rs = AUX + 2 * D_DIM;
  float* u = AUX + 3 * D_DIM;
  float* v = AUX + 4 * D_DIM;
  float* bbp = AUX + 5 * D_DIM;
  float* partials = AUX + 6 * D_DIM;   // 2048*3
  float* errp = AUX + 9 * D_DIM;
  unsigned* cnt = (unsigned*)(errp + 1);

  ot_init_kernel<<<1, 1, 0, stream>>>(errp, cnt);
  ot_rowstats_kernel<<<D_DIM / 8, 256, 0, stream>>>(theta, emb, sq, nrm);
  ot_gram_kernel<<<(D_DIM / 64) * (D_DIM / 64), 256, 0, stream>>>(theta, emb, sq, nrm, Kmat, Pnn);
  ot_rowsum_kernel<<<D_DIM / 8, 256, 0, stream>>>(Pnn, rs);
  ot_sinkhorn_kernel<<<D_DIM / 8, 256, 0, stream>>>(Kmat, u, v, bbp, errp, cnt);
  ot_kl_partial_kernel<<<D_DIM, 256, 0, stream>>>(Kmat, Pnn, rs, u, v, partials);
  ot_kl_final_kernel<<<1, 256, 0, stream>>>(partials, out);
}